// GraphAttentionCriticNetwork_35562329211476
// MI455X (gfx1250) — compile-verified
//
#include <hip/hip_runtime.h>

typedef float v2f __attribute__((ext_vector_type(2)));
typedef float v8f __attribute__((ext_vector_type(8)));
typedef int   vsi4 __attribute__((__vector_size__(4 * sizeof(int))));

// D = A(16x4,f32) * B(4x16,f32) + C(16x16,f32)  -- full fp32 matrix pipe
#define WMMA_F32(a, b, c) \
  __builtin_amdgcn_wmma_f32_16x16x4_f32(false, (a), false, (b), (short)0, (c), false, false)

#if defined(__AMDGCN__) && __has_builtin(__builtin_amdgcn_global_load_async_to_lds_b128) && \
    __has_builtin(__builtin_amdgcn_global_load_async_to_lds_b32) && \
    __has_builtin(__builtin_amdgcn_s_wait_asynccnt)
#define HAVE_ASYNC_LDS 1
#else
#define HAVE_ASYNC_LDS 0
#endif

#define GPTR(T) __attribute__((address_space(1))) T*
#define LPTR(T) __attribute__((address_space(3))) T*

__device__ __forceinline__ float lrelu(float x) { return x > 0.f ? x : 0.01f * x; }

__global__ __launch_bounds__(32)
void gac_value_kernel(const float* __restrict__ states,    // [256,16,64]
                      const float* __restrict__ policies,  // [256,16,8]
                      const float* __restrict__ actions,   // [256,16,8]
                      const float* __restrict__ W_emb,     // [128,72]
                      const float* __restrict__ W_att,     // [1,256]
                      const float* __restrict__ W1,        // [64,192]
                      const float* __restrict__ W2,        // [1,64]
                      float* __restrict__ out)             // value[65536] ++ ret_w[65536]
{
  constexpr int OBS = 64, D = 128, H = 64, I = 72;
  constexpr int RET_OFF = 256 * 16 * 16;

  const int b     = blockIdx.x;
  const int lane  = threadIdx.x;            // wave32: one wave per batch
  const int mrow  = lane & 15;              // A-frag row / B-frag col / C col
  const int khalf = (lane >> 4) * 2;        // K sub-offset per half-wave
  const int crow  = (lane >> 4) * 8;        // C row offset per half-wave

  __shared__ float S[16][80];      // [states | actions | policies] per node
  __shared__ float EA[16][128];    // embA raw -> tanhA
  __shared__ float EP[16][128];    // embP raw -> diffP = tanhP - tanhA
  __shared__ float AG[16][128];    // (w @ tanhA) / 16
  __shared__ float Wt[16][16];     // attention weights
  __shared__ float U[16][64];      // i-only part of hidden pre-activation
  __shared__ float VP[16][64];     // W1a . diffP[j]
  __shared__ float ssrc[16];
  __shared__ float sdst[16];
  __shared__ float W2s[64];

  // ---- stage inputs into LDS -------------------------------------------
  const float* st = states   + b * 16 * 64;
  const float* ac = actions  + b * 16 * 8;
  const float* po = policies + b * 16 * 8;

#if HAVE_ASYNC_LDS
  // CDNA5 async copy: global -> LDS direct, tracked by ASYNCcnt (no VGPR
  // round-trip). Row r of states lands at S[r][0..63]; row stride 80 floats
  // keeps every 4-float chunk 16B-aligned (80*4 = 320 bytes).
  #pragma unroll
  for (int i = 0; i < 8; ++i) {
    const int li = i * 32 + lane;             // 256 chunks of 4 floats
    const int r = li >> 4, c = (li & 15) << 2;
    __builtin_amdgcn_global_load_async_to_lds_b128(
        (GPTR(vsi4))(st + li * 4), (LPTR(vsi4))&S[r][c], 0, 0);
  }
  #pragma unroll
  for (int i0 = 0; i0 < 128; i0 += 32) {
    const int i = i0 + lane;
    __builtin_amdgcn_global_load_async_to_lds_b32(
        (GPTR(int))(ac + i), (LPTR(int))&S[i >> 3][64 + (i & 7)], 0, 0);
    __builtin_amdgcn_global_load_async_to_lds_b32(
        (GPTR(int))(po + i), (LPTR(int))&S[i >> 3][72 + (i & 7)], 0, 0);
  }
  W2s[lane]      = W2[lane];
  W2s[lane + 32] = W2[lane + 32];
  __builtin_amdgcn_s_wait_asynccnt(0);
#else
  for (int i = lane; i < 16 * 64; i += 32) S[i >> 6][i & 63] = st[i];
  for (int i = lane; i < 16 * 8; i += 32) {
    S[i >> 3][64 + (i & 7)] = ac[i];
    S[i >> 3][72 + (i & 7)] = po[i];
  }
  W2s[lane]      = W2[lane];
  W2s[lane + 32] = W2[lane + 32];
#endif
  __syncthreads();

  // ---- embA = oa @ W_emb^T, embP = op @ W_emb^T  (shared B fragments) ---
  #pragma unroll 1
  for (int n0 = 0; n0 < D; n0 += 16) {
    v8f accA = {}; v8f accP = {};
    const float* wrow = W_emb + (n0 + mrow) * I;   // B[k][n] = W_emb[n][k]
    #pragma unroll
    for (int k0 = 0; k0 < I; k0 += 4) {
      const int k = k0 + khalf;
      v2f bf; bf.x = wrow[k];      bf.y = wrow[k + 1];
      v2f aA; aA.x = S[mrow][k];   aA.y = S[mrow][k + 1];
      const int cp = (k < OBS) ? k : (k + 8);      // policy cols live at +8
      v2f aP; aP.x = S[mrow][cp];  aP.y = S[mrow][cp + 1];
      accA = WMMA_F32(aA, bf, accA);
      accP = WMMA_F32(aP, bf, accP);
    }
    #pragma unroll
    for (int r = 0; r < 8; r++) {
      EA[r + crow][n0 + mrow] = accA[r];
      EP[r + crow][n0 + mrow] = accP[r];
    }
  }
  __syncthreads();

  // ---- attention logits: s_src (lanes 0-15) / s_dst (lanes 16-31) ------
  {
    const int n = mrow;
    const float* wa = W_att + ((lane >> 4) ? D : 0);
    float s = 0.f;
    for (int d = 0; d < D; ++d) s += EA[n][d] * wa[d];
    if (lane < 16) ssrc[n] = s; else sdst[n] = s;
  }
  __syncthreads();

  // ---- softmax over neighbors j; emit ret_weight ------------------------
  if (lane < 16) {
    const int i = lane;
    float sc[16];
    float mx = -3.4e38f;
    #pragma unroll
    for (int j = 0; j < 16; j++) {
      float x = lrelu(ssrc[i] + sdst[j]);
      sc[j] = x;
      mx = fmaxf(mx, x);
    }
    float sum = 0.f;
    #pragma unroll
    for (int j = 0; j < 16; j++) { float e = __expf(sc[j] - mx); sc[j] = e; sum += e; }
    const float inv = 1.f / sum;
    #pragma unroll
    for (int j = 0; j < 16; j++) {
      const float w = sc[j] * inv;
      Wt[i][j] = w;
      out[RET_OFF + b * 256 + i * 16 + j] = w;
    }
  }

  // ---- tanh both embeddings; EP becomes diffP ---------------------------
  for (int idx = lane; idx < 16 * 128; idx += 32) {
    const int r = idx >> 7, c = idx & 127;
    const float ta = tanhf(EA[r][c]);
    const float tp = tanhf(EP[r][c]);
    EA[r][c] = ta;
    EP[r][c] = tp - ta;
  }
  __syncthreads();

  // ---- AG = (w @ tanhA) / 16  ------------------------------------------
  #pragma unroll 1
  for (int n0 = 0; n0 < D; n0 += 16) {
    v8f acc = {};
    #pragma unroll
    for (int k0 = 0; k0 < 16; k0 += 4) {
      const int k = k0 + khalf;
      v2f a;  a.x  = Wt[mrow][k];        a.y  = Wt[mrow][k + 1];
      v2f bf; bf.x = EA[k][n0 + mrow];   bf.y = EA[k + 1][n0 + mrow];
      acc = WMMA_F32(a, bf, acc);
    }
    #pragma unroll
    for (int r = 0; r < 8; r++) AG[r + crow][n0 + mrow] = acc[r] * 0.0625f;
  }
  __syncthreads();

  // ---- U = [states | AG] @ W1^T ;  VP = diffP @ W1a^T  (shared W1 rows) -
  #pragma unroll 1
  for (int n0 = 0; n0 < H; n0 += 16) {
    v8f accU = {}; v8f accV = {};
    const float* w1row = W1 + (n0 + mrow) * 192;
    #pragma unroll 8
    for (int k0 = 0; k0 < 192; k0 += 4) {
      const int k = k0 + khalf;
      v2f bu; bu.x = w1row[k]; bu.y = w1row[k + 1];
      v2f au;
      if (k < OBS) { au.x = S[mrow][k];        au.y = S[mrow][k + 1]; }
      else         { au.x = AG[mrow][k - 64];  au.y = AG[mrow][k - 63]; }
      accU = WMMA_F32(au, bu, accU);
      if (k0 < D) {  // VP K-range is 128
        v2f bv; bv.x = w1row[64 + k];  bv.y = w1row[64 + k + 1];
        v2f av; av.x = EP[mrow][k];    av.y = EP[mrow][k + 1];
        accV = WMMA_F32(av, bv, accV);
      }
    }
    #pragma unroll
    for (int r = 0; r < 8; r++) {
      U[r + crow][n0 + mrow]  = accU[r];
      VP[r + crow][n0 + mrow] = accV[r];
    }
  }
  __syncthreads();

  // ---- value[i,j] = W2 . leaky(U[i] + (w[i,j]/16) * VP[j]) --------------
  #pragma unroll
  for (int t = 0; t < 8; t++) {
    const int p = lane * 8 + t;         // 256 (i,j) pairs over 32 lanes
    const int i = p >> 4, j = p & 15;
    const float wj = Wt[i][j] * 0.0625f;
    float acc = 0.f;
    for (int h = 0; h < 64; h++) {
      acc += lrelu(U[i][h] + wj * VP[j][h]) * W2s[h];
    }
    out[b * 256 + p] = acc;
  }
}

extern "C" void kernel_launch(void* const* d_in, const int* in_sizes, int n_in,
                              void* d_out, int out_size, void* d_ws, size_t ws_size,
                              hipStream_t stream) {
  (void)in_sizes; (void)n_in; (void)out_size; (void)d_ws; (void)ws_size;
  const float* states   = (const float*)d_in[0];
  const float* policies = (const float*)d_in[1];
  const float* actions  = (const float*)d_in[2];
  const float* W_emb    = (const float*)d_in[3];
  const float* W_att    = (const float*)d_in[4];
  const float* W1       = (const float*)d_in[5];
  const float* W2       = (const float*)d_in[6];
  gac_value_kernel<<<256, 32, 0, stream>>>(states, policies, actions,
                                           W_emb, W_att, W1, W2, (float*)d_out);
}